// CustomLoss_90434831384972
// MI455X (gfx1250) — compile-verified
//
#include <hip/hip_runtime.h>

typedef __attribute__((ext_vector_type(2))) float v2f;
typedef __attribute__((ext_vector_type(8))) float v8f;

#define BB 32
#define NN 128
#define MM 128
#define DD 256

// LDS layout: cost (64KB) + p2/g2 (1KB) + 6 arrays of 130 dwords
#define SMEM_BYTES (65536 + 1024 + 6 * 130 * 4 + 64)

__global__ __launch_bounds__(256) void hungarian_fused_kernel(
    const float* __restrict__ P,   // [B, N, D]
    const float* __restrict__ G,   // [B, M, D]
    float* __restrict__ ws)        // [B] per-batch matched sums
{
  extern __shared__ char smem[];
  float* costS = (float*)smem;                         // 128*128 fp32 cost
  float* p2s   = (float*)(smem + 65536);               // 128
  float* g2s   = p2s + 128;                            // 128
  volatile float* uS    = (volatile float*)(g2s + 128); // 129 (pad 130)
  volatile float* vS    = uS + 130;
  volatile float* minvS = vS + 130;
  volatile int*   pS    = (volatile int*)(minvS + 130);
  volatile int*   wayS  = pS + 130;
  volatile int*   usedS = wayS + 130;

  const int b = blockIdx.x;
  const int t = threadIdx.x;
  const float* Pb = P + (size_t)b * NN * DD;
  const float* Gb = G + (size_t)b * MM * DD;

  // ---------------- Phase 0: squared row norms (256 threads, 1 row each) ---
  {
    const float* src = (t < NN) ? (Pb + t * DD) : (Gb + (t - NN) * DD);
    float acc = 0.f;
    #pragma unroll 4
    for (int d = 0; d < DD; d += 4) {
      float4 x = *(const float4*)(src + d);
      acc += x.x * x.x + x.y * x.y + x.z * x.z + x.w * x.w;
    }
    if (t < NN) p2s[t] = acc; else g2s[t - NN] = acc;
  }
  __syncthreads();

  // ---------------- Phase 1: cost tiles via fp32 WMMA into LDS ------------
  {
    const int wave = t >> 5;
    const int lane = t & 31;
    const int r  = lane & 15;   // row/col within tile
    const int kh = lane >> 4;   // K half-select (K pairs 0-1 vs 2-3)
    for (int tile = wave; tile < 64; tile += 8) {
      const int tm = tile >> 3;          // 16-row block of P
      const int tn = tile & 7;           // 16-row block of G
      const float* arow = Pb + (size_t)(tm * 16 + r) * DD;
      const float* brow = Gb + (size_t)(tn * 16 + r) * DD;
      v8f acc = {};
      #pragma unroll 8
      for (int k = 0; k < DD; k += 4) {
        const int kb = k + 2 * kh;
        v2f a  = *(const v2f*)(arow + kb);   // A: 16x4 fp32 fragment
        v2f bf = *(const v2f*)(brow + kb);   // B: 4x16 fp32 fragment (G^T)
        acc = __builtin_amdgcn_wmma_f32_16x16x4_f32(
            false, a, false, bf, (short)0, acc, false, false);
      }
      // C layout: VGPR e -> M = e + 8*kh, N = r (lane half)
      #pragma unroll
      for (int e = 0; e < 8; e++) {
        const int gi = tm * 16 + e + 8 * kh;
        const int gj = tn * 16 + r;
        costS[gi * MM + gj] =
            (p2s[gi] + g2s[gj] - 2.0f * acc[e]) * (1.0f / (float)DD);
      }
    }
  }
  __syncthreads();

  // ---------------- Phase 2: Jonker-Volgenant Hungarian on wave 0 ---------
  if (t >= 32) return;
  const int lane = t;
  const float INF = __builtin_huge_valf();

  for (int j = lane; j <= MM; j += 32) { uS[j] = 0.f; vS[j] = 0.f; pS[j] = 0; }

  for (int i = 1; i <= NN; i++) {
    if (lane == 0) { pS[0] = i; usedS[0] = 0; }
    #pragma unroll
    for (int s = 0; s < 4; s++) {
      const int j = 1 + lane + 32 * s;
      minvS[j] = INF;
      usedS[j] = 0;
    }
    int j0 = 0;
    for (;;) {
      if (lane == 0) usedS[j0] = 1;
      const int i0 = pS[j0];
      const float ui0 = uS[i0];
      float best = INF;
      int bj = 129;
      #pragma unroll
      for (int s = 0; s < 4; s++) {
        const int j = 1 + lane + 32 * s;
        if (!usedS[j]) {
          const float cur = costS[(i0 - 1) * MM + (j - 1)] - ui0 - vS[j];
          float mv = minvS[j];
          if (cur < mv) { mv = cur; minvS[j] = cur; wayS[j] = j0; }
          if (mv < best || (mv == best && j < bj)) { best = mv; bj = j; }
        }
      }
      // butterfly argmin, lowest index on ties (matches np.argmin)
      #pragma unroll
      for (int off = 16; off >= 1; off >>= 1) {
        const float ob = __shfl_xor(best, off, 32);
        const int   oj = __shfl_xor(bj,   off, 32);
        if (ob < best || (ob == best && oj < bj)) { best = ob; bj = oj; }
      }
      const float delta = best;
      #pragma unroll
      for (int s = 0; s < 4; s++) {
        const int j = 1 + lane + 32 * s;
        if (usedS[j]) { uS[pS[j]] += delta; vS[j] -= delta; }
        else          { minvS[j]  -= delta; }
      }
      if (lane == 0) { uS[pS[0]] += delta; vS[0] -= delta; }
      j0 = bj;
      if (pS[j0] == 0) break;
    }
    if (lane == 0) {           // augment along "way" chain
      int j = j0;
      while (j) {
        const int jn = wayS[j];
        pS[j] = pS[jn];
        j = jn;
      }
    }
  }

  // ---------------- Phase 3: matched-cost sum for this batch --------------
  float s = 0.f;
  #pragma unroll
  for (int q = 0; q < 4; q++) {
    const int j = 1 + lane + 32 * q;
    const int pi = pS[j];
    if (pi > 0) s += costS[(pi - 1) * MM + (j - 1)];
  }
  #pragma unroll
  for (int off = 16; off >= 1; off >>= 1) s += __shfl_xor(s, off, 32);
  if (lane == 0) ws[b] = s;
}

__global__ void finalize_kernel(const float* __restrict__ ws,
                                float* __restrict__ out) {
  if (threadIdx.x == 0 && blockIdx.x == 0) {
    float acc = 0.f;
    for (int b = 0; b < BB; b++) acc += ws[b];
    out[0] = acc * (1.0f / ((float)NN * (float)BB));  // WEIGHT = 1
  }
}

extern "C" void kernel_launch(void* const* d_in, const int* in_sizes, int n_in,
                              void* d_out, int out_size, void* d_ws, size_t ws_size,
                              hipStream_t stream) {
  (void)in_sizes; (void)n_in; (void)out_size; (void)ws_size;
  const float* P = (const float*)d_in[0];
  const float* G = (const float*)d_in[1];
  float* out = (float*)d_out;
  float* ws  = (float*)d_ws;

  // Raise dynamic-LDS cap above 64KB (host-side attribute, capture-safe).
  (void)hipFuncSetAttribute(reinterpret_cast<const void*>(hungarian_fused_kernel),
                            hipFuncAttributeMaxDynamicSharedMemorySize,
                            SMEM_BYTES);

  hungarian_fused_kernel<<<dim3(BB), dim3(256), SMEM_BYTES, stream>>>(P, G, ws);
  finalize_kernel<<<1, 32, 0, stream>>>(ws, out);
}